// L2Normalize_59219009077998
// MI455X (gfx1250) — compile-verified
//
#include <hip/hip_runtime.h>

// ---------------------------------------------------------------------------
// L2 row-normalization, rows of 512 f32. Bandwidth-bound (1.07 GB @ 23.3 TB/s).
// One wave32 per row; async global->LDS double-buffered pipeline (CDNA5 path);
// cross-lane sum via V_WMMA_F32_16X16X4_F32 against an all-ones B matrix.
// ---------------------------------------------------------------------------

#define WAVES_PER_BLOCK 8
#define ROWS_PER_WAVE   8
#define ROW_LEN         512               // floats per row
#define THREADS         (WAVES_PER_BLOCK * 32)

typedef __attribute__((ext_vector_type(2))) float v2f;
typedef __attribute__((ext_vector_type(4))) float f4;
typedef __attribute__((ext_vector_type(8))) float v8f;
typedef __attribute__((ext_vector_type(4))) int   v4i;

#ifndef __has_builtin
#define __has_builtin(x) 0
#endif

#if __has_builtin(__builtin_amdgcn_global_load_async_to_lds_b128) && \
    __has_builtin(__builtin_amdgcn_s_wait_asynccnt)
#define USE_ASYNC_LDS 1
#else
#define USE_ASYNC_LDS 0
#endif

#if __has_builtin(__builtin_amdgcn_wmma_f32_16x16x4_f32)
#define USE_WMMA_REDUCE 1
#else
#define USE_WMMA_REDUCE 0
#endif

#if USE_ASYNC_LDS
// Builtin signature (from clang diagnostic): arg0 = int4 addrspace(1)* (global),
// arg1 = int4 addrspace(3)* (LDS), then imm offset, imm cpol.
typedef __attribute__((address_space(1))) v4i g_v4i;
typedef __attribute__((address_space(3))) v4i l_v4i;

// Issue 4 async b128 loads covering one 2KB row: lane l moves bytes
// [l*16 + k*512, l*16 + k*512 + 16) for k = 0..3 (imm offset applies to both
// the global and LDS addresses per the ISA async-load semantics).
__device__ __forceinline__ void async_load_row(const float* grow, float* lbuf, int lane) {
  char* gp = (char*)grow + (size_t)lane * 16;
  char* lp = (char*)lbuf + (size_t)lane * 16;
  g_v4i* g = (g_v4i*)gp;
  l_v4i* l = (l_v4i*)lp;
  __builtin_amdgcn_global_load_async_to_lds_b128(g, l, 0,    0);
  __builtin_amdgcn_global_load_async_to_lds_b128(g, l, 512,  0);
  __builtin_amdgcn_global_load_async_to_lds_b128(g, l, 1024, 0);
  __builtin_amdgcn_global_load_async_to_lds_b128(g, l, 1536, 0);
}
#endif

__device__ __forceinline__ float sq_acc(f4 a, float s) {
  s = fmaf(a.x, a.x, s);
  s = fmaf(a.y, a.y, s);
  s = fmaf(a.z, a.z, s);
  s = fmaf(a.w, a.w, s);
  return s;
}

// Sum one f32 value across all 32 lanes; result valid in every lane.
// WMMA variant: A(16x4) holds the 32 partials in each lane's first A-VGPR and
// zeros in the second; B(4x16) = all ones => D[i][j] = rowsum_i(A). Each lane's
// 8 D-VGPRs hold 8 distinct row-sums (M=0..7 for lanes 0-15, M=8..15 for lanes
// 16-31), so 7 adds + one xor-16 shuffle yields sum over all A entries = sum of
// all 32 partials, regardless of A's exact (lane,vgpr)->(m,k) mapping.
__device__ __forceinline__ float wave_allsum(float v) {
#if USE_WMMA_REDUCE
  v2f a; a[0] = v;    a[1] = 0.0f;
  v2f b; b[0] = 1.0f; b[1] = 1.0f;
  v8f c = {};
  v8f d = __builtin_amdgcn_wmma_f32_16x16x4_f32(false, a, false, b,
                                                (short)0, c, false, false);
  float s = ((d[0] + d[1]) + (d[2] + d[3])) + ((d[4] + d[5]) + (d[6] + d[7]));
  s += __shfl_xor(s, 16, 32);
  return s;
#else
  #pragma unroll
  for (int m = 16; m > 0; m >>= 1) v += __shfl_xor(v, m, 32);
  return v;
#endif
}

__global__ __launch_bounds__(THREADS)
void l2norm_rows_kernel(const float* __restrict__ in, float* __restrict__ out, int nrows) {
  const int lane = (int)(threadIdx.x & 31u);
  const int wave = (int)(threadIdx.x >> 5u);
  const long long rowBase =
      ((long long)blockIdx.x * WAVES_PER_BLOCK + wave) * ROWS_PER_WAVE;

#if USE_ASYNC_LDS
  __shared__ float lds[WAVES_PER_BLOCK * 2 * ROW_LEN];   // 32 KB: 2KB x 2 bufs x 8 waves
  float* const bufA = &lds[wave * 2 * ROW_LEN];
  float* const bufB = bufA + ROW_LEN;
  if (rowBase < nrows)                                    // wave-uniform
    async_load_row(in + rowBase * ROW_LEN, bufA, lane);
#endif

#pragma unroll 1
  for (int i = 0; i < ROWS_PER_WAVE; ++i) {
    const long long row = rowBase + i;
    if (row >= nrows) break;                              // wave-uniform

#if USE_ASYNC_LDS
    const bool hasNext = (i + 1 < ROWS_PER_WAVE) && (row + 1 < nrows); // uniform
    if (hasNext)
      async_load_row(in + (row + 1) * ROW_LEN, ((i & 1) ? bufA : bufB), lane);
    if (hasNext) __builtin_amdgcn_s_wait_asynccnt(4);     // row i's 4 ops done (in-order)
    else         __builtin_amdgcn_s_wait_asynccnt(0);
    __asm__ volatile("" ::: "memory");
    const f4* __restrict__ src = (const f4*)((i & 1) ? bufB : bufA);
#else
    const f4* __restrict__ src = (const f4*)(in + row * ROW_LEN);
#endif

    // Lane l handles float4 chunks l, l+32, l+64, l+96 (coalesced / conflict-free).
    f4 x0 = src[lane];
    f4 x1 = src[lane + 32];
    f4 x2 = src[lane + 64];
    f4 x3 = src[lane + 96];

    float ssq = 0.0f;
    ssq = sq_acc(x0, ssq);
    ssq = sq_acc(x1, ssq);
    ssq = sq_acc(x2, ssq);
    ssq = sq_acc(x3, ssq);

    const float tot = wave_allsum(ssq);
    const float inv = 1.0f / __builtin_sqrtf(tot);

    f4* dst = (f4*)(out + row * ROW_LEN);
    __builtin_nontemporal_store(x0 * inv, dst + lane);        // streamed output:
    __builtin_nontemporal_store(x1 * inv, dst + lane + 32);   // never re-read
    __builtin_nontemporal_store(x2 * inv, dst + lane + 64);
    __builtin_nontemporal_store(x3 * inv, dst + lane + 96);
  }
}

extern "C" void kernel_launch(void* const* d_in, const int* in_sizes, int n_in,
                              void* d_out, int out_size, void* d_ws, size_t ws_size,
                              hipStream_t stream) {
  (void)n_in; (void)out_size; (void)d_ws; (void)ws_size;
  const float* feats = (const float*)d_in[0];
  float* out = (float*)d_out;
  const int nrows = in_sizes[0] / ROW_LEN;                  // 64*4096 = 262144
  const int rowsPerBlock = WAVES_PER_BLOCK * ROWS_PER_WAVE; // 64
  const int grid = (nrows + rowsPerBlock - 1) / rowsPerBlock;
  l2norm_rows_kernel<<<grid, THREADS, 0, stream>>>(feats, out, nrows);
}